// PACE2dLayer_39651138077499
// MI455X (gfx1250) — compile-verified
//
#include <hip/hip_runtime.h>
#include <math.h>
#include <stdint.h>

// ---------------------------------------------------------------------------
// PACE2dLayer for MI455X (gfx1250): every stage expressed as fp32 WMMA GEMMs.
// Truncated rfft2/irfft2 are done as DFT-matrix GEMMs (only 80x70 modes kept).
// LDS staging uses CDNA5 GLOBAL_LOAD_ASYNC_TO_LDS_B128 (ASYNCcnt-tracked).
// ---------------------------------------------------------------------------

typedef __attribute__((ext_vector_type(2))) float v2f;
typedef __attribute__((ext_vector_type(8))) float v8f;

#define TWO_PI 6.28318530717958647692f

__device__ __forceinline__ int lane_id() { return (int)(threadIdx.x & 31u); }

__device__ __forceinline__ v8f wmma4(v2f a, v2f b, v8f c) {
  // D = A(16x4,f32) x B(4x16,f32) + C(16x16,f32)
  return __builtin_amdgcn_wmma_f32_16x16x4_f32(false, a, false, b, (short)0, c,
                                               false, false);
}

// CDNA5 async bulk copy: 16B global -> LDS, tracked by ASYNCcnt.
__device__ __forceinline__ void async_copy_b128(void* lds, const void* g) {
  uint32_t loff = (uint32_t)(uintptr_t)lds;  // wave-relative LDS byte address
  asm volatile("global_load_async_to_lds_b128 %0, %1, off" ::"v"(loff), "v"(g)
               : "memory");
}
__device__ __forceinline__ void wait_async0() {
  asm volatile("s_wait_asynccnt 0x0" ::: "memory");
}

// A fragment from global: A[m][k], row stride rs, with K bound (zero pad).
__device__ __forceinline__ v2f ldA_g(const float* A, long long rs, int m0,
                                     int k0, int kmax) {
  int l = lane_id();
  long long m = m0 + (l & 15);
  int k = k0 + ((l >> 4) << 1);
  v2f a;
  a.x = (k < kmax) ? A[m * rs + k] : 0.f;
  a.y = (k + 1 < kmax) ? A[m * rs + k + 1] : 0.f;
  return a;
}

// B fragment from global: B[k][n], row stride rs, with N and K bounds.
__device__ __forceinline__ v2f ldB_g(const float* Bm, long long rs, int k0,
                                     int n0, int nmax, int kmax) {
  int l = lane_id();
  int n = n0 + (l & 15);
  int k = k0 + ((l >> 4) << 1);
  bool nin = n < nmax;
  v2f b;
  b.x = (nin && k < kmax) ? Bm[(long long)k * rs + n] : 0.f;
  b.y = (nin && k + 1 < kmax) ? Bm[(long long)(k + 1) * rs + n] : 0.f;
  return b;
}

// ----------------------------- problem sizes -------------------------------
// B=8, C=64, H=256, W=256, M1=40 (x2 halves -> 80 H-modes), M2=70 W-modes.
#define NB 8
#define NC 64
#define NH 256
#define NW 256
#define MH 80           // kept H modes (40 low + 40 high)
#define MW 70           // kept W (rfft) modes
#define HW 65536        // NH*NW
#define NPIX (NB * HW)  // 524288
#define NROW (NB * NC * NH)  // 131072 rows of length NW

// ------------------------- twiddle table builders --------------------------
__global__ void k_tw_w(float* CW, float* SW) {  // [NW][MW]
  int i = blockIdx.x * blockDim.x + threadIdx.x;
  if (i >= NW * MW) return;
  int w = i / MW, kw = i % MW;
  float th = TWO_PI * (float)((w * kw) & 255) * (1.0f / 256.0f);
  CW[i] = cosf(th);
  SW[i] = -sinf(th);
}

__global__ void k_tw_h(float* EHr, float* EHi) {  // [MH][NH], e^{-i t}
  int i = blockIdx.x * blockDim.x + threadIdx.x;
  if (i >= MH * NH) return;
  int m = i / NH, h = i % NH;
  int kh = (m < 40) ? m : (m + 176);  // 216 + (m-40)
  float th = TWO_PI * (float)((kh * h) & 255) * (1.0f / 256.0f);
  EHr[i] = cosf(th);
  EHi[i] = -sinf(th);
}

__global__ void k_tw_ih(float* EIHr, float* EIHi) {  // [NH][MH], e^{+i t}/65536
  int i = blockIdx.x * blockDim.x + threadIdx.x;
  if (i >= NH * MH) return;
  int h = i / MH, m = i % MH;
  int kh = (m < 40) ? m : (m + 176);
  float th = TWO_PI * (float)((kh * h) & 255) * (1.0f / 256.0f);
  const float s = 1.0f / 65536.0f;
  EIHr[i] = cosf(th) * s;
  EIHi[i] = sinf(th) * s;
}

__global__ void k_tw_iw(float* CIW, float* SIW) {  // [MW][NW], hermitian x2
  int i = blockIdx.x * blockDim.x + threadIdx.x;
  if (i >= MW * NW) return;
  int kw = i / NW, w = i % NW;
  float c = (kw == 0) ? 1.0f : 2.0f;
  float th = TWO_PI * (float)((kw * w) & 255) * (1.0f / 256.0f);
  CIW[i] = c * cosf(th);
  SIW[i] = -c * sinf(th);
}

// ------------------ spectral weight transpose to [mode][o][i] --------------
__global__ void k_wt(const float* w1r, const float* w1i, const float* w2r,
                     const float* w2i, float* t1r, float* t1i, float* t2r,
                     float* t2i) {
  long long idx = (long long)blockIdx.x * blockDim.x + threadIdx.x;
  const long long N = 64LL * 64 * 40 * 70;
  if (idx >= N) return;
  int i = (int)(idx / (64 * 40 * 70));
  int rem = (int)(idx % (64 * 40 * 70));
  int o = rem / (40 * 70);
  int mkw = rem % (40 * 70);  // m*70+kw
  long long d = (long long)mkw * 4096 + o * 64 + i;
  t1r[d] = w1r[idx];
  t1i[d] = w1i[idx];
  t2r[d] = w2r[idx];
  t2i[d] = w2i[idx];
}

// ----------------------- stage 1: conv1x1 + exact GELU ---------------------
__global__ __launch_bounds__(256) void k_proj_gelu(const float* __restrict__ x,
                                                   const float* __restrict__ w,
                                                   const float* __restrict__ bias,
                                                   float* __restrict__ u) {
  __shared__ float Wl[64 * 64];  // w[o][i] row-major (already A-layout)
  int tid = threadIdx.x;
  for (int c = tid; c < 1024; c += 256)  // 1024 x 16B async chunks
    async_copy_b128(&Wl[c * 4], w + c * 4);
  wait_async0();
  __syncthreads();
  int wv = tid >> 5, l = lane_id();
  int mt = wv >> 1, nt = wv & 1;
  long long p = (long long)blockIdx.x * 32 + nt * 16 + (l & 15);
  long long bb = p >> 16, hw = p & 65535;
  const float* xb = x + bb * (64LL * HW) + hw;
  v8f acc = {};
#pragma unroll
  for (int k0 = 0; k0 < 64; k0 += 4) {
    int k = k0 + ((l >> 4) << 1);
    int mo = mt * 16 + (l & 15);
    v2f a;
    a.x = Wl[mo * 64 + k];
    a.y = Wl[mo * 64 + k + 1];
    v2f b;
    b.x = xb[(long long)k * HW];
    b.y = xb[(long long)(k + 1) * HW];
    acc = wmma4(a, b, acc);
  }
  float* ub = u + bb * (64LL * HW) + hw;
#pragma unroll
  for (int r = 0; r < 8; ++r) {
    int o = mt * 16 + r + ((l >> 4) << 3);
    float v = acc[r] + bias[o];
    v = 0.5f * v * (1.0f + erff(v * 0.70710678118654752f));
    ub[(long long)o * HW] = v;
  }
}

// --------------------- stage 2: DFT along W (256 -> 70) --------------------
__global__ __launch_bounds__(160) void k_dft_w(const float* __restrict__ u,
                                               const float* __restrict__ CW,
                                               const float* __restrict__ SW,
                                               float* __restrict__ Yr,
                                               float* __restrict__ Yi) {
  __shared__ float Au[16 * 256];  // 16 contiguous rows of u (16KB)
  long long row0 = (long long)blockIdx.x * 16;
  int tid = threadIdx.x;
  const float* ubase = u + row0 * 256;
  for (int c = tid; c < 1024; c += 160)  // 1024 x 16B async chunks
    async_copy_b128(&Au[c * 4], ubase + c * 4);
  wait_async0();
  __syncthreads();
  int wv = tid >> 5, l = lane_id();  // 5 waves = 5 N-tiles
  int n0 = wv * 16;
  v8f ar = {}, ai = {};
  for (int k0 = 0; k0 < 256; k0 += 4) {
    int k = k0 + ((l >> 4) << 1);
    v2f a;
    a.x = Au[(l & 15) * 256 + k];
    a.y = Au[(l & 15) * 256 + k + 1];
    v2f bc = ldB_g(CW, MW, k0, n0, MW, 256);
    v2f bs = ldB_g(SW, MW, k0, n0, MW, 256);
    ar = wmma4(a, bc, ar);
    ai = wmma4(a, bs, ai);
  }
  int n = n0 + (l & 15);
  if (n < MW) {
    int mb = (l >> 4) << 3;
#pragma unroll
    for (int r = 0; r < 8; ++r) {
      Yr[(row0 + mb + r) * MW + n] = ar[r];
      Yi[(row0 + mb + r) * MW + n] = ai[r];
    }
  }
}

// --------------------- stage 3: DFT along H (256 -> 80) --------------------
__global__ __launch_bounds__(256) void k_dft_h(const float* __restrict__ Yr,
                                               const float* __restrict__ Yi,
                                               const float* __restrict__ EHr,
                                               const float* __restrict__ EHi,
                                               float* __restrict__ Zr,
                                               float* __restrict__ Zi) {
  int bc = blockIdx.x;  // 0..511
  const float* Br_ = Yr + (long long)bc * NH * MW;
  const float* Bi_ = Yi + (long long)bc * NH * MW;
  float* Zr_ = Zr + (long long)bc * MH * MW;
  float* Zi_ = Zi + (long long)bc * MH * MW;
  int wv = threadIdx.x >> 5, l = lane_id();
  for (int t = wv; t < 25; t += 8) {  // 5 M-tiles x 5 N-tiles
    int m0 = (t / 5) * 16, n0 = (t % 5) * 16;
    v8f ar = {}, ai = {};
    for (int k0 = 0; k0 < 256; k0 += 4) {
      v2f Ar = ldA_g(EHr, NH, m0, k0, NH);
      v2f Ai = ldA_g(EHi, NH, m0, k0, NH);
      v2f Br = ldB_g(Br_, MW, k0, n0, MW, NH);
      v2f Bi = ldB_g(Bi_, MW, k0, n0, MW, NH);
      v2f nBi = -Bi;
      ar = wmma4(Ar, Br, ar);
      ar = wmma4(Ai, nBi, ar);
      ai = wmma4(Ar, Bi, ai);
      ai = wmma4(Ai, Br, ai);
    }
    int n = n0 + (l & 15);
    if (n < MW) {
      int mb = m0 + ((l >> 4) << 3);
#pragma unroll
      for (int r = 0; r < 8; ++r) {
        Zr_[(long long)(mb + r) * MW + n] = ar[r];
        Zi_[(long long)(mb + r) * MW + n] = ai[r];
      }
    }
  }
}

// ------------- stage 4: per-mode complex channel mix (64x64 GEMM) ----------
// out[b,o] = sum_i Z[b,i] * W[i,o];  N = [8 batch-Re | 8 batch-Im] packed.
__global__ __launch_bounds__(128) void k_specmix(
    const float* __restrict__ Zr, const float* __restrict__ Zi,
    const float* __restrict__ t1r, const float* __restrict__ t1i,
    const float* __restrict__ t2r, const float* __restrict__ t2i,
    float* __restrict__ Z2r, float* __restrict__ Z2i) {
  __shared__ float Wr[4096], Wi[4096];
  __shared__ float Zs[16][64];  // [8 b-Re | 8 b-Im][i]
  int mode = blockIdx.x;        // m*70 + kw, m in [0,80)
  int m = mode / MW;
  const float* sr = (m < 40) ? t1r : t2r;
  const float* si = (m < 40) ? t1i : t2i;
  long long woff = (long long)((m < 40) ? mode : (mode - 40 * MW)) * 4096;
  int tid = threadIdx.x;
  for (int c = tid; c < 1024; c += 128) {  // async weight staging (32KB)
    async_copy_b128(&Wr[c * 4], sr + woff + c * 4);
    async_copy_b128(&Wi[c * 4], si + woff + c * 4);
  }
  const long long zi_str = (long long)MH * MW;  // i stride in Z
  const long long zb_str = 64LL * MH * MW;      // b stride in Z
  for (int j = tid; j < 1024; j += 128) {       // gather this mode's Z once
    int n = j >> 6, i = j & 63;
    const float* src = (n < 8) ? Zr : Zi;
    Zs[n][i] = src[(long long)(n & 7) * zb_str + (long long)i * zi_str + mode];
  }
  wait_async0();
  __syncthreads();
  int wv = tid >> 5, l = lane_id();
  int m0 = wv * 16;  // o-tile
  int n = l & 15;
  int bcol = n & 7;
  bool isIm = n >= 8;
  int n2 = isIm ? (n - 8) : (n + 8);
  float sgn = isIm ? 1.0f : -1.0f;
  v8f acc = {};
#pragma unroll 4
  for (int k0 = 0; k0 < 64; k0 += 4) {
    int k = k0 + ((l >> 4) << 1);
    int mo = m0 + (l & 15);
    v2f Ar, Ai, B1, B2;
    Ar.x = Wr[mo * 64 + k];
    Ar.y = Wr[mo * 64 + k + 1];
    Ai.x = Wi[mo * 64 + k];
    Ai.y = Wi[mo * 64 + k + 1];
    B1.x = Zs[n][k];          // [Zr | Zi]
    B1.y = Zs[n][k + 1];
    B2.x = sgn * Zs[n2][k];   // [-Zi | Zr]
    B2.y = sgn * Zs[n2][k + 1];
    acc = wmma4(Ar, B1, acc);  // [Wr*Zr | Wr*Zi]
    acc = wmma4(Ai, B2, acc);  // [-Wi*Zi | Wi*Zr]  -> [Re | Im]
  }
  float* Zout = (isIm ? Z2i : Z2r) + (long long)bcol * zb_str + mode;
  int mb = m0 + ((l >> 4) << 3);
#pragma unroll
  for (int r = 0; r < 8; ++r) Zout[(long long)(mb + r) * zi_str] = acc[r];
}

// ------------------- stage 5: inverse DFT along H (80 -> 256) --------------
__global__ __launch_bounds__(256) void k_idft_h(const float* __restrict__ Z2r,
                                                const float* __restrict__ Z2i,
                                                const float* __restrict__ EIHr,
                                                const float* __restrict__ EIHi,
                                                float* __restrict__ Y2r,
                                                float* __restrict__ Y2i) {
  int bc = blockIdx.x;
  const float* Br_ = Z2r + (long long)bc * MH * MW;
  const float* Bi_ = Z2i + (long long)bc * MH * MW;
  float* Or = Y2r + (long long)bc * NH * MW;
  float* Oi = Y2i + (long long)bc * NH * MW;
  int wv = threadIdx.x >> 5, l = lane_id();
  for (int t = wv; t < 80; t += 8) {  // 16 M-tiles x 5 N-tiles
    int m0 = (t / 5) * 16, n0 = (t % 5) * 16;
    v8f ar = {}, ai = {};
    for (int k0 = 0; k0 < MH; k0 += 4) {
      v2f Ar = ldA_g(EIHr, MH, m0, k0, MH);
      v2f Ai = ldA_g(EIHi, MH, m0, k0, MH);
      v2f Br = ldB_g(Br_, MW, k0, n0, MW, MH);
      v2f Bi = ldB_g(Bi_, MW, k0, n0, MW, MH);
      v2f nBi = -Bi;
      ar = wmma4(Ar, Br, ar);
      ar = wmma4(Ai, nBi, ar);
      ai = wmma4(Ar, Bi, ai);
      ai = wmma4(Ai, Br, ai);
    }
    int n = n0 + (l & 15);
    if (n < MW) {
      int mb = m0 + ((l >> 4) << 3);
#pragma unroll
      for (int r = 0; r < 8; ++r) {
        Or[(long long)(mb + r) * MW + n] = ar[r];
        Oi[(long long)(mb + r) * MW + n] = ai[r];
      }
    }
  }
}

// ------------------- stage 6: inverse rDFT along W (70 -> 256) -------------
__global__ __launch_bounds__(256) void k_idft_w(const float* __restrict__ Y2r,
                                                const float* __restrict__ Y2i,
                                                const float* __restrict__ CIW,
                                                const float* __restrict__ SIW,
                                                float* __restrict__ f1) {
  __shared__ float Ars[16 * 72], Ais[16 * 72];  // K padded 70 -> 72
  long long row0 = (long long)blockIdx.x * 16;
  int tid = threadIdx.x;
  for (int j = tid; j < 16 * 72; j += 256) {
    int rr = j / 72, kk = j % 72;
    float vr = 0.f, vi = 0.f;
    if (kk < MW) {
      vr = Y2r[(row0 + rr) * MW + kk];
      vi = Y2i[(row0 + rr) * MW + kk];
    }
    Ars[j] = vr;
    Ais[j] = vi;
  }
  __syncthreads();
  int wv = tid >> 5, l = lane_id();
  for (int nt = wv; nt < 16; nt += 8) {
    int n0 = nt * 16;
    v8f acc = {};
    for (int k0 = 0; k0 < 72; k0 += 4) {
      int k = k0 + ((l >> 4) << 1);
      v2f Ar, Ai;
      Ar.x = Ars[(l & 15) * 72 + k];
      Ar.y = Ars[(l & 15) * 72 + k + 1];
      Ai.x = Ais[(l & 15) * 72 + k];
      Ai.y = Ais[(l & 15) * 72 + k + 1];
      v2f Bc = ldB_g(CIW, NW, k0, n0, NW, MW);
      v2f Bs = ldB_g(SIW, NW, k0, n0, NW, MW);
      acc = wmma4(Ar, Bc, acc);
      acc = wmma4(Ai, Bs, acc);
    }
    int n = n0 + (l & 15);
    int mb = (l >> 4) << 3;
#pragma unroll
    for (int r = 0; r < 8; ++r) f1[(row0 + mb + r) * 256 + n] = acc[r];
  }
}

// ------------- stage 7: out = (W2 f1 + b2) * sigmoid(W3 u + b3) ------------
__global__ __launch_bounds__(256) void k_final(
    const float* __restrict__ f1, const float* __restrict__ u,
    const float* __restrict__ w2, const float* __restrict__ b2,
    const float* __restrict__ w3, const float* __restrict__ b3,
    float* __restrict__ out) {
  __shared__ float W2l[4096], W3l[4096];
  int tid = threadIdx.x;
  for (int c = tid; c < 1024; c += 256) {  // async weight staging (32KB)
    async_copy_b128(&W2l[c * 4], w2 + c * 4);
    async_copy_b128(&W3l[c * 4], w3 + c * 4);
  }
  wait_async0();
  __syncthreads();
  int wv = tid >> 5, l = lane_id();
  int mt = wv >> 1, nt = wv & 1;
  long long p = (long long)blockIdx.x * 32 + nt * 16 + (l & 15);
  long long bb = p >> 16, hw = p & 65535;
  const float* fb = f1 + bb * (64LL * HW) + hw;
  const float* ub = u + bb * (64LL * HW) + hw;
  v8f a2 = {}, a3 = {};
#pragma unroll
  for (int k0 = 0; k0 < 64; k0 += 4) {
    int k = k0 + ((l >> 4) << 1);
    int mo = mt * 16 + (l & 15);
    v2f A2, A3, Bf, Bu;
    A2.x = W2l[mo * 64 + k];
    A2.y = W2l[mo * 64 + k + 1];
    A3.x = W3l[mo * 64 + k];
    A3.y = W3l[mo * 64 + k + 1];
    Bf.x = fb[(long long)k * HW];
    Bf.y = fb[(long long)(k + 1) * HW];
    Bu.x = ub[(long long)k * HW];
    Bu.y = ub[(long long)(k + 1) * HW];
    a2 = wmma4(A2, Bf, a2);
    a3 = wmma4(A3, Bu, a3);
  }
  float* ob = out + bb * (64LL * HW) + hw;
#pragma unroll
  for (int r = 0; r < 8; ++r) {
    int o = mt * 16 + r + ((l >> 4) << 3);
    float v2 = a2[r] + b2[o];
    float v3 = a3[r] + b3[o];
    float g = 1.0f / (1.0f + expf(-v3));
    ob[(long long)o * HW] = v2 * g;
  }
}

// ---------------------------------------------------------------------------
extern "C" void kernel_launch(void* const* d_in, const int* in_sizes, int n_in,
                              void* d_out, int out_size, void* d_ws,
                              size_t ws_size, hipStream_t stream) {
  const float* x = (const float*)d_in[0];
  const float* w_proj1 = (const float*)d_in[1];
  const float* b_proj1 = (const float*)d_in[2];
  const float* w1r = (const float*)d_in[3];
  const float* w1i = (const float*)d_in[4];
  const float* w2r = (const float*)d_in[5];
  const float* w2i = (const float*)d_in[6];
  const float* w_conv2 = (const float*)d_in[7];
  const float* b_conv2 = (const float*)d_in[8];
  const float* w_conv3 = (const float*)d_in[9];
  const float* b_conv3 = (const float*)d_in[10];
  float* out = (float*)d_out;

  // workspace layout (float units)
  float* ws = (float*)d_ws;
  size_t off = 0;
  const size_t N_u = (size_t)NB * NC * HW;          // 33,554,432
  const size_t N_Y = (size_t)NB * NC * NH * MW;     //  9,175,040
  const size_t N_Z = (size_t)NB * NC * MH * MW;     //  2,867,200
  const size_t N_Wt = (size_t)64 * 64 * 40 * 70;    // 11,468,800
  float* u = ws + off; off += N_u;
  float* f1 = ws + off; off += N_u;
  float* Yr = ws + off; off += N_Y;
  float* Yi = ws + off; off += N_Y;
  float* Zr = ws + off; off += N_Z;
  float* Zi = ws + off; off += N_Z;
  float* Z2r = ws + off; off += N_Z;
  float* Z2i = ws + off; off += N_Z;
  float* t1r = ws + off; off += N_Wt;
  float* t1i = ws + off; off += N_Wt;
  float* t2r = ws + off; off += N_Wt;
  float* t2i = ws + off; off += N_Wt;
  float* CW = ws + off; off += (size_t)NW * MW;
  float* SW = ws + off; off += (size_t)NW * MW;
  float* EHr = ws + off; off += (size_t)MH * NH;
  float* EHi = ws + off; off += (size_t)MH * NH;
  float* EIHr = ws + off; off += (size_t)NH * MH;
  float* EIHi = ws + off; off += (size_t)NH * MH;
  float* CIW = ws + off; off += (size_t)MW * NW;
  float* SIW = ws + off; off += (size_t)MW * NW;

  // twiddle tables
  k_tw_w<<<(NW * MW + 255) / 256, 256, 0, stream>>>(CW, SW);
  k_tw_h<<<(MH * NH + 255) / 256, 256, 0, stream>>>(EHr, EHi);
  k_tw_ih<<<(NH * MH + 255) / 256, 256, 0, stream>>>(EIHr, EIHi);
  k_tw_iw<<<(MW * NW + 255) / 256, 256, 0, stream>>>(CIW, SIW);

  // spectral weight transpose to [mode][o][i]
  k_wt<<<(unsigned)((N_Wt + 255) / 256), 256, 0, stream>>>(
      w1r, w1i, w2r, w2i, t1r, t1i, t2r, t2i);

  // pipeline
  k_proj_gelu<<<NPIX / 32, 256, 0, stream>>>(x, w_proj1, b_proj1, u);
  k_dft_w<<<NROW / 16, 160, 0, stream>>>(u, CW, SW, Yr, Yi);
  k_dft_h<<<NB * NC, 256, 0, stream>>>(Yr, Yi, EHr, EHi, Zr, Zi);
  k_specmix<<<MH * MW, 128, 0, stream>>>(Zr, Zi, t1r, t1i, t2r, t2i, Z2r, Z2i);
  k_idft_h<<<NB * NC, 256, 0, stream>>>(Z2r, Z2i, EIHr, EIHi, Yr, Yi);
  k_idft_w<<<NROW / 16, 256, 0, stream>>>(Yr, Yi, CIW, SIW, f1);
  k_final<<<NPIX / 32, 256, 0, stream>>>(f1, u, w_conv2, b_conv2, w_conv3,
                                         b_conv3, out);
}